// VectorQuantizer_68444598829798
// MI455X (gfx1250) — compile-verified
//
#include <hip/hip_runtime.h>

typedef __attribute__((ext_vector_type(2))) float v2f;
typedef __attribute__((ext_vector_type(8))) float v8f;

#define K_CODES 1024
#define CDIM    64
#define HW      1024
#define NTOT    32768   // 32 batches * 1024 spatial positions

// ---------- prep 0a: transpose codebook emb[k][c] -> embT[c][k] (L2-resident) ----------
__global__ void vq_prep_transpose(const float* __restrict__ emb, float* __restrict__ embT) {
    int tid = blockIdx.x * blockDim.x + threadIdx.x;   // 0 .. 65535
    int k = tid >> 6;
    int c = tid & 63;
    embT[c * K_CODES + k] = emb[tid];
}

// ---------- prep 0b: norms[k] = ||e_k||^2 (reads embT column-coalesced) ----------
__global__ void vq_prep_norms(const float* __restrict__ embT, float* __restrict__ norms) {
    int k = blockIdx.x * blockDim.x + threadIdx.x;     // 0 .. 1023
    float s = 0.f;
#pragma unroll
    for (int c = 0; c < CDIM; ++c) {
        float v = embT[c * K_CODES + k];
        s = fmaf(v, v, s);
    }
    norms[k] = s;
}

// ---------- main: fused GEMM (f32 WMMA) + argmin over K ----------
// Each wave32 handles 2 M-strips of 16 vectors (32 rows) over all 1024 codes.
// Block = 256 threads = 8 waves = 256 rows; grid = 32768/256 = 128 blocks.
__global__ __launch_bounds__(256) void vq_argmin_wmma(
        const float* __restrict__ z,       // [32][64][1024] channel-major
        const float* __restrict__ embT,    // [64][1024]
        const float* __restrict__ norms,   // [1024]
        int* __restrict__ idx_out) {       // [32768]
    const int tid  = threadIdx.x;
    const int wave = tid >> 5;
    const int lane = tid & 31;
    const int lm   = lane & 15;   // position within half-wave
    const int half = lane >> 4;   // 0: K 0-1 / M 0-7 side, 1: K 2-3 / M 8-15 side

    const int n0 = blockIdx.x * 256 + wave * 32;

    // ---- preload A: two 16x64 strips of z_flat into registers ----
    // A-layout (16x4 f32): lane<16 -> (M=lm, K=0|1), lane>=16 -> (M=lm, K=2|3)
    v2f a[2][16];
#pragma unroll
    for (int s = 0; s < 2; ++s) {
        int n  = n0 + s * 16 + lm;
        int b  = n >> 10;
        int hw = n & 1023;
        const float* zb = z + (size_t)b * CDIM * HW + hw;
#pragma unroll
        for (int cg = 0; cg < 16; ++cg) {
            int c = cg * 4 + half * 2;
            a[s][cg].x = zb[(size_t)c * HW];
            a[s][cg].y = zb[(size_t)(c + 1) * HW];
        }
    }

    float mn[2][8];
    int   mi[2][8];
#pragma unroll
    for (int s = 0; s < 2; ++s)
#pragma unroll
        for (int r = 0; r < 8; ++r) { mn[s][r] = 3.4028235e38f; mi[s][r] = 0; }

    // ---- sweep all 64 k-tiles of 16 codes ----
    for (int kt = 0; kt < K_CODES / 16; ++kt) {
        const int k0 = kt * 16;
        v8f acc0 = {};
        v8f acc1 = {};
#pragma unroll
        for (int cg = 0; cg < 16; ++cg) {
            int c = cg * 4 + half * 2;
            v2f bv;
            bv.x = embT[(size_t)c * K_CODES + k0 + lm];
            bv.y = embT[(size_t)(c + 1) * K_CODES + k0 + lm];
            acc0 = __builtin_amdgcn_wmma_f32_16x16x4_f32(
                       false, a[0][cg], false, bv, (short)0, acc0, false, false);
            acc1 = __builtin_amdgcn_wmma_f32_16x16x4_f32(
                       false, a[1][cg], false, bv, (short)0, acc1, false, false);
        }
        const int   kk = k0 + lm;
        const float nr = norms[kk];
        // branchless running argmin: keeps EXEC all-ones for the next WMMA loop
#pragma unroll
        for (int r = 0; r < 8; ++r) {
            float s0 = fmaf(-2.f, acc0[r], nr);
            bool  l0 = s0 < mn[0][r];
            mn[0][r] = l0 ? s0 : mn[0][r];
            mi[0][r] = l0 ? kk : mi[0][r];
            float s1 = fmaf(-2.f, acc1[r], nr);
            bool  l1 = s1 < mn[1][r];
            mn[1][r] = l1 ? s1 : mn[1][r];
            mi[1][r] = l1 ? kk : mi[1][r];
        }
    }

    // ---- reduce across the 16 lanes of each half (C/D tile: N striped over lanes) ----
#pragma unroll
    for (int s = 0; s < 2; ++s) {
#pragma unroll
        for (int r = 0; r < 8; ++r) {
            float v = mn[s][r];
            int   i = mi[s][r];
#pragma unroll
            for (int m = 8; m >= 1; m >>= 1) {
                float ov = __shfl_xor(v, m, 32);
                int   oi = __shfl_xor(i, m, 32);
                if (ov < v || (ov == v && oi < i)) { v = ov; i = oi; }
            }
            if (lm == 0) {
                int mrow = half * 8 + r;           // lanes16-31 hold M = r+8
                idx_out[n0 + s * 16 + mrow] = i;
            }
        }
    }
}

// ---------- gather: quantized[b][c][hw] = emb[idx[b][hw]][c], coalesced writes ----------
__global__ void vq_gather(const float* __restrict__ emb, const int* __restrict__ idx,
                          float* __restrict__ outq) {
    int bc = blockIdx.x;        // b*64 + c
    int b  = bc >> 6;
    int c  = bc & 63;
    int t  = threadIdx.x;
#pragma unroll
    for (int it = 0; it < 4; ++it) {
        int hw = it * 256 + t;
        int k  = idx[b * HW + hw];
        outq[(size_t)bc * HW + hw] = emb[k * CDIM + c];
    }
}

extern "C" void kernel_launch(void* const* d_in, const int* in_sizes, int n_in,
                              void* d_out, int out_size, void* d_ws, size_t ws_size,
                              hipStream_t stream) {
    const float* z   = (const float*)d_in[0];   // [32,64,32,32] f32
    const float* emb = (const float*)d_in[1];   // [1024,64] f32

    float* outq    = (float*)d_out;                         // 2,097,152 f32
    int*   idx_out = (int*)(outq + (size_t)NTOT * CDIM);    // 32,768 i32

    float* embT  = (float*)d_ws;                 // 64*1024 f32 = 256 KB
    float* norms = embT + CDIM * K_CODES;        // 1024 f32

    vq_prep_transpose<<<256, 256, 0, stream>>>(emb, embT);
    vq_prep_norms<<<K_CODES / 256, 256, 0, stream>>>(embT, norms);
    vq_argmin_wmma<<<NTOT / 256, 256, 0, stream>>>(z, embT, norms, idx_out);
    vq_gather<<<32 * CDIM, 256, 0, stream>>>(emb, idx_out, outq);
}